// SecondaryStatistic_17652315586730
// MI455X (gfx1250) — compile-verified
//
#include <hip/hip_runtime.h>

#define B_      64
#define HW_     3136
#define C_      128
#define KC      32            // spatial rows per LDS chunk
#define LDSK    (KC + 2)      // pad to 34 dwords: 8B alignment + conflict-free b64 reads
#define NCHUNK  (HW_ / KC)    // 98
#define LBUF    (C_ * LDSK)   // floats per chunk buffer

typedef __attribute__((ext_vector_type(2))) float v2f;
typedef __attribute__((ext_vector_type(8))) float v8f;

// One workgroup = (batch b, 16-row output block r). 8 waves; wave w computes the
// 16x16 tile (rows 16r..16r+15, cols 16w..16w+15) of cov[b] via f32 WMMA.
// Double-buffered LDS pipeline: one barrier per chunk; next chunk's global
// loads overlap the current chunk's WMMAs.
__global__ __launch_bounds__(256) void secstat_cov_wmma(const float* __restrict__ x,
                                                        float* __restrict__ out) {
  __shared__ float lds[2 * LBUF];    // 34,816 B : two chunks, channel-major [ch][row]
  __shared__ float ssum_sh[8 * C_];  //  4,096 B : per-wave channel-sum partials

  const int tid  = threadIdx.x;
  const int lane = tid & 31;
  const int wave = tid >> 5;
  const int b    = blockIdx.x;      // batch
  const int r    = blockIdx.y;      // output row block

  // WMMA f32 16x16x4 fragment addressing (ISA 7.12.2):
  //  A (16x4, M=row-block channels): lane L<16 -> M=L, K={0,1}; L>=16 -> M=L-16, K={2,3}
  //  B (4x16, N=col-block channels): lane L<16 -> N=L, K={0,1}; L>=16 -> N=L-16, K={2,3}
  // With LDS channel-major [ch*LDSK + row], each fragment is one aligned b64 load.
  const int laneN = lane & 15;
  const int laneK = (lane >> 4) << 1;                 // 0 or 2
  const int aBase = (16 * r    + laneN) * LDSK + laneK;
  const int bBase = (16 * wave + laneN) * LDSK + laneK;

  const float4* __restrict__ xg4 = (const float4*)(x + (size_t)b * HW_ * C_);

  // Copy mapping (fixed per thread): flat = (i*256 + tid)*4
  //  -> channels [4*(tid&31), +4), row = 8*i + wave
  const int ch0 = (lane << 2);

  v8f acc = {0.f, 0.f, 0.f, 0.f, 0.f, 0.f, 0.f, 0.f};
  float s0 = 0.f, s1 = 0.f, s2 = 0.f, s3 = 0.f;      // per-thread channel sums

  // Prologue: stage chunk 0 into buffer 0.
  {
#pragma unroll
    for (int i = 0; i < 4; ++i) {
      const float4 v = xg4[i * 256 + tid];            // coalesced b128
      const int row = (i << 3) + wave;
      lds[(ch0 + 0) * LDSK + row] = v.x;
      lds[(ch0 + 1) * LDSK + row] = v.y;
      lds[(ch0 + 2) * LDSK + row] = v.z;
      lds[(ch0 + 3) * LDSK + row] = v.w;
      s0 += v.x; s1 += v.y; s2 += v.z; s3 += v.w;
    }
  }

  for (int chunk = 0; chunk < NCHUNK; ++chunk) {
    __syncthreads();                                  // copy(chunk) done; buf safe to read
    const float* Lc = lds + (chunk & 1) * LBUF;
#pragma unroll
    for (int k = 0; k < KC; k += 4) {
      const v2f a  = *(const v2f*)&Lc[aBase + k];     // ds_load_b64
      const v2f bf = *(const v2f*)&Lc[bBase + k];     // ds_load_b64
      acc = __builtin_amdgcn_wmma_f32_16x16x4_f32(false, a, false, bf,
                                                  (short)0, acc, false, false);
    }
    if (chunk + 1 < NCHUNK) {                         // stage next chunk (overlaps WMMA)
      float* Ln = lds + ((chunk + 1) & 1) * LBUF;
      const int base4 = (chunk + 1) * KC * (C_ / 4);
#pragma unroll
      for (int i = 0; i < 4; ++i) {
        const float4 v = xg4[base4 + i * 256 + tid];  // coalesced b128
        const int row = (i << 3) + wave;
        Ln[(ch0 + 0) * LDSK + row] = v.x;
        Ln[(ch0 + 1) * LDSK + row] = v.y;
        Ln[(ch0 + 2) * LDSK + row] = v.z;
        Ln[(ch0 + 3) * LDSK + row] = v.w;
        s0 += v.x; s1 += v.y; s2 += v.z; s3 += v.w;
      }
    }
  }

  // Deterministic channel-sum reduction: 8 wave-partials -> 128 totals.
  __syncthreads();
  ssum_sh[wave * C_ + ch0 + 0] = s0;
  ssum_sh[wave * C_ + ch0 + 1] = s1;
  ssum_sh[wave * C_ + ch0 + 2] = s2;
  ssum_sh[wave * C_ + ch0 + 3] = s3;
  __syncthreads();
  if (tid < C_) {
    float s = 0.f;
#pragma unroll
    for (int w = 0; w < 8; ++w) s += ssum_sh[w * C_ + tid];
    ssum_sh[tid] = s;                                 // totals live in [0,128)
  }
  __syncthreads();

  // cov = (G - s_i*s_j/hw) / (hw-1)
  const float invHW = 1.0f / (float)HW_;
  const float invN1 = 1.0f / (float)(HW_ - 1);
  const int  colC = 16 * wave + laneN;
  const float sj  = ssum_sh[colC];
#pragma unroll
  for (int v = 0; v < 8; ++v) {
    const int rowC = 16 * r + v + ((lane >> 4) << 3); // C/D layout: VGPR v, M=v(+8)
    const float si = ssum_sh[rowC];
    const float cv = (acc[v] - si * sj * invHW) * invN1;
    out[((size_t)b * C_ + rowC) * C_ + colC] = cv;
  }
}

extern "C" void kernel_launch(void* const* d_in, const int* in_sizes, int n_in,
                              void* d_out, int out_size, void* d_ws, size_t ws_size,
                              hipStream_t stream) {
  (void)in_sizes; (void)n_in; (void)d_ws; (void)ws_size; (void)out_size;
  const float* x = (const float*)d_in[0];
  float* out = (float*)d_out;
  dim3 grid(B_, C_ / 16);   // 64 batches x 8 row blocks
  dim3 block(256);          // 8 waves
  secstat_cov_wmma<<<grid, block, 0, stream>>>(x, out);
}